// HSEGNN_81844896793189
// MI455X (gfx1250) — compile-verified
//
#include <hip/hip_runtime.h>
#include <math.h>

// ---------------------------------------------------------------------------
// HSEGNN layer for MI455X (gfx1250, wave32).  All small matmuls are mapped to
// V_WMMA_F32_16X16X4_F32 (exact fp32).  One wave == one 16-row batch tile.
// ---------------------------------------------------------------------------

typedef float v2f __attribute__((ext_vector_type(2)));
typedef float v8f __attribute__((ext_vector_type(8)));

constexpr int NN = 50000;   // particles
constexpr int EE = 800000;  // edges

#define DEV __device__ __forceinline__

DEV float siluf(float x) { return x / (1.0f + __expf(-x)); }

// --- wave-level GEMM: D[16x16] += A[16xK] * B[Kx16] over nk k-steps of 4 ----
// A staged row-major in LDS with row stride `strideK` floats.
// B pre-packed in workspace as [ntile][kstep][lane][2] matching the assumed
// fragment layout: lanes 0-15 hold K pair {4k,4k+1}, lanes 16-31 {4k+2,4k+3}.
DEV v8f wave_gemm(const float* As, int strideK, const float* Bp, int ntile, int nk) {
  const int lane = threadIdx.x & 31;
  const int row  = lane & 15;
  const int koff = (lane >> 4) << 1;
  v8f acc = {};
  const float* B = Bp + (ntile * nk) * 64;
  for (int ks = 0; ks < nk; ++ks) {
    const float* ap = As + row * strideK + (ks << 2) + koff;
    v2f a; a.x = ap[0]; a.y = ap[1];
    const float* bp = B + (ks << 6) + (lane << 1);
    v2f b; b.x = bp[0]; b.y = bp[1];
    acc = __builtin_amdgcn_wmma_f32_16x16x4_f32(false, a, false, b,
                                                (short)0, acc, false, false);
  }
  return acc;
}

// C/D fragment -> LDS row-major [16][stride], column offset col0.
// VGPR i: lanes0-15 = (M=i, N=lane), lanes16-31 = (M=i+8, N=lane-16).
DEV void store_tile(v8f acc, float* out, int stride, int col0) {
  const int lane = threadIdx.x & 31;
  const int n  = lane & 15;
  const int m0 = (lane >> 4) << 3;
#pragma unroll
  for (int i = 0; i < 8; ++i) out[(m0 + i) * stride + col0 + n] = acc[i];
}

// Gated combine of a (s_out=48,v_out=16) TP:  scalars -> os[16][32],
// gated vectors -> ov[16][16][3].  Optional per-node aggregate add.
DEV void gate_combine(const float* Ssrc /*[16][48]*/, const float* Tsrc /*[16][16]*/,
                      const float* PQsrc /*3x[16][16]*/, const float* bias,
                      const float* esc, float* os, float* ov,
                      const float* aggs, const float* aggv, int n0) {
  const int lane = threadIdx.x & 31;
  for (int idx = lane; idx < 512; idx += 32) {
    int r = idx >> 5, j = idx & 31;
    float v = siluf(Ssrc[r * 48 + j] + bias[j]);
    if (aggs) v += aggs[(n0 + r) * 32 + j];
    os[idx] = v;
  }
  for (int idx = lane; idx < 768; idx += 32) {
    int r = idx / 48, rem = idx - r * 48;
    int k = rem / 3, c = rem - k * 3;
    float g = siluf(Ssrc[r * 48 + 32 + k] + bias[32 + k]);
    float v = (Tsrc[r * 16 + k] * esc[r * 4 + 1 + c] +
               PQsrc[c * 256 + r * 16 + k]) * g;
    if (aggv) v += aggv[(n0 + r) * 48 + rem];
    ov[idx] = v;
  }
  __syncthreads();
}

// Full small TP with s_in=32, v_in=16 (cp2/m2/u2).  Inputs: xs at Ast[0..511]
// ([16][32]), xv at Ast[512..1279] ([16][16][3]).  Results: out_s at Ast[1280]
// (stride so, nt1 n-tiles), t at Ast[2048], pq at Ast[2304 + c*256].
DEV void tp_small(const float* W1, const float* W2, const float* W3,
                  float* Ast, float* Dst, const float* esc, int nt1, int so) {
  const int lane = threadIdx.x & 31;
  // A1 [16][48] = [xs*a0 (32), dot (16)]
  for (int idx = lane; idx < 768; idx += 32) {
    int r = idx / 48, k = idx - r * 48;
    float val;
    if (k < 32) val = Ast[r * 32 + k] * esc[r * 4];
    else {
      const float* xv = Ast + 512 + r * 48 + (k - 32) * 3;
      val = xv[0] * esc[r*4+1] + xv[1] * esc[r*4+2] + xv[2] * esc[r*4+3];
    }
    Dst[idx] = val;
  }
  __syncthreads();
  for (int nt = 0; nt < nt1; ++nt) {
    v8f d = wave_gemm(Dst, 48, W1, nt, 12);
    store_tile(d, Ast + 1280, so, nt * 16);
  }
  __syncthreads();
  { v8f t = wave_gemm(Ast, 32, W2, 0, 8); store_tile(t, Ast + 2048, 16, 0); }
  __syncthreads();
  // A3 [48][32], row = c*16+r : [xv_c*a0 (16), cross_c (16)]
  for (int idx = lane; idx < 1536; idx += 32) {
    int rr = idx >> 5, j = idx & 31;
    int c = rr >> 4, r = rr & 15;
    int v = (j < 16) ? j : j - 16;
    const float* xv = Ast + 512 + r * 48 + v * 3;
    float val;
    if (j < 16) val = xv[c] * esc[r * 4];
    else {
      int c1 = (c == 2) ? 0 : c + 1;
      int c2 = (c == 0) ? 2 : c - 1;
      val = xv[c1] * esc[r*4+1+c2] - xv[c2] * esc[r*4+1+c1];
    }
    Dst[idx] = val;
  }
  __syncthreads();
  for (int c = 0; c < 3; ++c) {
    v8f p = wave_gemm(Dst + c * 512, 32, W3, 0, 8);
    store_tile(p, Ast + 2304 + c * 256, 16, 0);
  }
  __syncthreads();
}

// ---------------------------------------------------------------------------
// Weight packing: compose [Wa;Wb;pad] (Kp x Np padded) into B-fragment layout.
__global__ void pack_kernel(const float* Wa, const float* Wb, int Ka, int Kb,
                            int Nout, int Kp, int Np, float* dst) {
  int idx = blockIdx.x * blockDim.x + threadIdx.x;
  int nk = Kp >> 2;
  int total = (Np >> 4) * nk * 32;
  if (idx >= total) return;
  int lane = idx & 31;
  int rest = idx >> 5;
  int ks = rest % nk;
  int nt = rest / nk;
  int n  = (nt << 4) + (lane & 15);
  int k0 = (ks << 2) + ((lane >> 4) << 1);
  float v0 = 0.f, v1 = 0.f;
  if (n < Nout) {
    int k = k0;
    v0 = (k < Ka) ? Wa[k * Nout + n] : ((k - Ka < Kb) ? Wb[(k - Ka) * Nout + n] : 0.f);
    k = k0 + 1;
    v1 = (k < Ka) ? Wa[k * Nout + n] : ((k - Ka < Kb) ? Wb[(k - Ka) * Nout + n] : 0.f);
  }
  dst[((nt * nk + ks) * 32 + lane) * 2 + 0] = v0;
  dst[((nt * nk + ks) * 32 + lane) * 2 + 1] = v1;
}

__global__ void zero_kernel(float* p, int n) {
  int i = blockIdx.x * blockDim.x + threadIdx.x;
  if (i < n) p[i] = 0.f;
}

// ---------------------------------------------------------------------------
struct EdgeArgs {
  const float *x_s, *x_v;
  const int* edge;                     // [2E]: first E = src, next E = dst
  const float *ea0, *eav, *amf;
  const float *W1a, *W2a, *W3a, *ba;   // m1 packed weights + raw bias
  const float *W1b, *W2b, *W3b, *bb;   // m2
  float *agg_s, *agg_v;
};

__global__ __launch_bounds__(32) void edge_kernel(EdgeArgs A) {
  __shared__ float Ast[3072];
  __shared__ float Dst[1792];
  __shared__ float esc[64];
  const int lane = threadIdx.x;
  const int e0 = blockIdx.x * 16;

  if (lane < 16) {
    int e = e0 + lane;
    esc[lane*4+0] = A.ea0[e];
    esc[lane*4+1] = A.eav[e*3+0];
    esc[lane*4+2] = A.eav[e*3+1];
    esc[lane*4+3] = A.eav[e*3+2];
  }
  __syncthreads();

  // ===== m1: s_in=65 v_in=32 =====
  // A1 [16][100] = [xs*a0 (65), dot (32), pad]
  for (int idx = lane; idx < 1600; idx += 32) {
    int r = idx / 100, k = idx - r * 100;
    int e = e0 + r;
    int si = A.edge[e], di = A.edge[EE + e];
    float val = 0.f;
    if (k < 65) {
      float xs;
      if (k < 32)      xs = A.x_s[di * 32 + k];
      else if (k < 64) xs = A.x_s[si * 32 + (k - 32)];
      else             xs = A.amf[e];
      val = xs * esc[r * 4];
    } else if (k < 97) {
      int v = k - 65;
      const float* xv = (v < 16) ? (A.x_v + (di * 16 + v) * 3)
                                 : (A.x_v + (si * 16 + (v - 16)) * 3);
      val = xv[0]*esc[r*4+1] + xv[1]*esc[r*4+2] + xv[2]*esc[r*4+3];
    }
    Ast[r * 100 + k] = val;
  }
  __syncthreads();
  for (int nt = 0; nt < 3; ++nt) {
    v8f d = wave_gemm(Ast, 100, A.W1a, nt, 25);
    store_tile(d, Dst, 48, nt * 16);
  }
  __syncthreads();
  // A2 [16][68] = [xs (65), pad]
  for (int idx = lane; idx < 1088; idx += 32) {
    int r = idx / 68, k = idx - r * 68;
    int e = e0 + r;
    float val = 0.f;
    if (k < 65) {
      int si = A.edge[e], di = A.edge[EE + e];
      if (k < 32)      val = A.x_s[di * 32 + k];
      else if (k < 64) val = A.x_s[si * 32 + (k - 32)];
      else             val = A.amf[e];
    }
    Ast[r * 68 + k] = val;
  }
  __syncthreads();
  { v8f t = wave_gemm(Ast, 68, A.W2a, 0, 17); store_tile(t, Dst + 768, 16, 0); }
  __syncthreads();
  // A3 [48][64], row = c*16+r : [xv_c*a0 (32), cross_c (32)]
  for (int idx = lane; idx < 3072; idx += 32) {
    int rr = idx >> 6, j = idx & 63;
    int c = rr >> 4, r = rr & 15;
    int e = e0 + r;
    int si = A.edge[e], di = A.edge[EE + e];
    int v = (j < 32) ? j : j - 32;
    const float* xv = (v < 16) ? (A.x_v + (di * 16 + v) * 3)
                               : (A.x_v + (si * 16 + (v - 16)) * 3);
    float val;
    if (j < 32) val = xv[c] * esc[r * 4];
    else {
      int c1 = (c == 2) ? 0 : c + 1;
      int c2 = (c == 0) ? 2 : c - 1;
      val = xv[c1] * esc[r*4+1+c2] - xv[c2] * esc[r*4+1+c1];
    }
    Ast[rr * 64 + j] = val;
  }
  __syncthreads();
  for (int c = 0; c < 3; ++c) {
    v8f p = wave_gemm(Ast + c * 1024, 64, A.W3a, 0, 16);
    store_tile(p, Dst + 1024 + c * 256, 16, 0);
  }
  __syncthreads();
  gate_combine(Dst, Dst + 768, Dst + 1024, A.ba, esc, Ast, Ast + 512,
               nullptr, nullptr, 0);

  // ===== m2 =====
  tp_small(A.W1b, A.W2b, A.W3b, Ast, Dst, esc, 3, 48);

  // ===== gate + scatter-add =====
  for (int idx = lane; idx < 512; idx += 32) {
    int r = idx >> 5, j = idx & 31;
    int di = A.edge[EE + e0 + r];
    atomicAdd(&A.agg_s[di * 32 + j], siluf(Ast[1280 + r * 48 + j] + A.bb[j]));
  }
  for (int idx = lane; idx < 768; idx += 32) {
    int r = idx / 48, rem = idx - r * 48;
    int k = rem / 3, c = rem - k * 3;
    int di = A.edge[EE + e0 + r];
    float g = siluf(Ast[1280 + r * 48 + 32 + k] + A.bb[32 + k]);
    float v = (Ast[2048 + r * 16 + k] * esc[r * 4 + 1 + c] +
               Ast[2304 + c * 256 + r * 16 + k]) * g;
    atomicAdd(&A.agg_v[di * 48 + rem], v);
  }
}

// ---------------------------------------------------------------------------
struct NodeArgs {
  const float *x_s, *x_v, *xc_s, *xc_v;
  const int* il_parent;
  const float *eila0, *eilav, *amf_il, *na0, *nav;
  const float *c1W1, *c1W2, *c1W3, *c1b;
  const float *c2W1, *c2W2, *c2W3, *c2b;
  const float *u1W1, *u1W2, *u1W3, *u1b;
  const float *u2W1, *u2W2, *u2W3, *u2b;
  const float *agg_s, *agg_v;          // aliases of out (accumulated by edges)
  float *out_s, *out_v;
};

__global__ __launch_bounds__(32) void node_kernel(NodeArgs A) {
  __shared__ float Ast[3072];
  __shared__ float Dst[1792];
  __shared__ float esc[64];
  __shared__ float nesc[64];
  const int lane = threadIdx.x;
  const int n0 = blockIdx.x * 16;

  if (lane < 16) {
    int n = n0 + lane;
    esc[lane*4+0]  = A.eila0[n];
    esc[lane*4+1]  = A.eilav[n*3+0];
    esc[lane*4+2]  = A.eilav[n*3+1];
    esc[lane*4+3]  = A.eilav[n*3+2];
    nesc[lane*4+0] = A.na0[n];
    nesc[lane*4+1] = A.nav[n*3+0];
    nesc[lane*4+2] = A.nav[n*3+1];
    nesc[lane*4+3] = A.nav[n*3+2];
  }
  __syncthreads();

  // ===== cp1: s_in=65 v_in=32 (node + parent cell) =====
  for (int idx = lane; idx < 1600; idx += 32) {
    int r = idx / 100, k = idx - r * 100;
    int n = n0 + r;
    int p = A.il_parent[n];
    float val = 0.f;
    if (k < 65) {
      float xs;
      if (k < 32)      xs = A.x_s[n * 32 + k];
      else if (k < 64) xs = A.xc_s[p * 32 + (k - 32)];
      else             xs = A.amf_il[n];
      val = xs * esc[r * 4];
    } else if (k < 97) {
      int v = k - 65;
      const float* xv = (v < 16) ? (A.x_v + (n * 16 + v) * 3)
                                 : (A.xc_v + (p * 16 + (v - 16)) * 3);
      val = xv[0]*esc[r*4+1] + xv[1]*esc[r*4+2] + xv[2]*esc[r*4+3];
    }
    Ast[r * 100 + k] = val;
  }
  __syncthreads();
  for (int nt = 0; nt < 3; ++nt) {
    v8f d = wave_gemm(Ast, 100, A.c1W1, nt, 25);
    store_tile(d, Dst, 48, nt * 16);
  }
  __syncthreads();
  for (int idx = lane; idx < 1088; idx += 32) {
    int r = idx / 68, k = idx - r * 68;
    int n = n0 + r;
    float val = 0.f;
    if (k < 65) {
      int p = A.il_parent[n];
      if (k < 32)      val = A.x_s[n * 32 + k];
      else if (k < 64) val = A.xc_s[p * 32 + (k - 32)];
      else             val = A.amf_il[n];
    }
    Ast[r * 68 + k] = val;
  }
  __syncthreads();
  { v8f t = wave_gemm(Ast, 68, A.c1W2, 0, 17); store_tile(t, Dst + 768, 16, 0); }
  __syncthreads();
  for (int idx = lane; idx < 3072; idx += 32) {
    int rr = idx >> 6, j = idx & 63;
    int c = rr >> 4, r = rr & 15;
    int n = n0 + r;
    int p = A.il_parent[n];
    int v = (j < 32) ? j : j - 32;
    const float* xv = (v < 16) ? (A.x_v + (n * 16 + v) * 3)
                               : (A.xc_v + (p * 16 + (v - 16)) * 3);
    float val;
    if (j < 32) val = xv[c] * esc[r * 4];
    else {
      int c1 = (c == 2) ? 0 : c + 1;
      int c2 = (c == 0) ? 2 : c - 1;
      val = xv[c1] * esc[r*4+1+c2] - xv[c2] * esc[r*4+1+c1];
    }
    Ast[rr * 64 + j] = val;
  }
  __syncthreads();
  for (int c = 0; c < 3; ++c) {
    v8f p = wave_gemm(Ast + c * 1024, 64, A.c1W3, 0, 16);
    store_tile(p, Dst + 1024 + c * 256, 16, 0);
  }
  __syncthreads();
  gate_combine(Dst, Dst + 768, Dst + 1024, A.c1b, esc, Ast, Ast + 512,
               nullptr, nullptr, 0);

  // ===== cp2, then msg = gate(cp2) + agg =====
  tp_small(A.c2W1, A.c2W2, A.c2W3, Ast, Dst, esc, 3, 48);
  gate_combine(Ast + 1280, Ast + 2048, Ast + 2304, A.c2b, esc, Ast, Ast + 512,
               A.agg_s, A.agg_v, n0);

  // ===== u1: s_in=64 v_in=32, inputs = [x, msg], attr = (na0, nav) =====
  for (int idx = lane; idx < 1536; idx += 32) {
    int r = idx / 96, k = idx - r * 96;
    int n = n0 + r;
    float val;
    if (k < 32)       val = A.x_s[n * 32 + k] * nesc[r * 4];
    else if (k < 64)  val = Ast[r * 32 + (k - 32)] * nesc[r * 4];
    else {
      int v = k - 64;
      const float* xv = (v < 16) ? (A.x_v + (n * 16 + v) * 3)
                                 : (Ast + 512 + r * 48 + (v - 16) * 3);
      val = xv[0]*nesc[r*4+1] + xv[1]*nesc[r*4+2] + xv[2]*nesc[r*4+3];
    }
    Dst[r * 96 + k] = val;
  }
  __syncthreads();
  for (int nt = 0; nt < 3; ++nt) {
    v8f d = wave_gemm(Dst, 96, A.u1W1, nt, 24);
    store_tile(d, Ast + 1280, 48, nt * 16);
  }
  __syncthreads();
  for (int idx = lane; idx < 1024; idx += 32) {
    int r = idx >> 6, k = idx & 63;
    int n = n0 + r;
    Dst[idx] = (k < 32) ? A.x_s[n * 32 + k] : Ast[r * 32 + (k - 32)];
  }
  __syncthreads();
  { v8f t = wave_gemm(Dst, 64, A.u1W2, 0, 16); store_tile(t, Ast + 2048, 16, 0); }
  __syncthreads();
  for (int c = 0; c < 3; ++c) {
    for (int idx = lane; idx < 1024; idx += 32) {
      int r = idx >> 6, j = idx & 63;
      int n = n0 + r;
      int v = (j < 32) ? j : j - 32;
      const float* xv = (v < 16) ? (A.x_v + (n * 16 + v) * 3)
                                 : (Ast + 512 + r * 48 + (v - 16) * 3);
      float val;
      if (j < 32) val = xv[c] * nesc[r * 4];
      else {
        int c1 = (c == 2) ? 0 : c + 1;
        int c2 = (c == 0) ? 2 : c - 1;
        val = xv[c1] * nesc[r*4+1+c2] - xv[c2] * nesc[r*4+1+c1];
      }
      Dst[idx] = val;
    }
    __syncthreads();
    v8f p = wave_gemm(Dst, 64, A.u1W3, 0, 16);
    store_tile(p, Ast + 2304 + c * 256, 16, 0);
    __syncthreads();
  }
  gate_combine(Ast + 1280, Ast + 2048, Ast + 2304, A.u1b, nesc, Ast, Ast + 512,
               nullptr, nullptr, 0);

  // ===== u2 (no gate) + residual write =====
  tp_small(A.u2W1, A.u2W2, A.u2W3, Ast, Dst, nesc, 2, 32);
  for (int idx = lane; idx < 512; idx += 32) {
    int r = idx >> 5, j = idx & 31;
    int n = n0 + r;
    A.out_s[n * 32 + j] = A.x_s[n * 32 + j] + Ast[1280 + r * 32 + j] + A.u2b[j];
  }
  for (int idx = lane; idx < 768; idx += 32) {
    int r = idx / 48, rem = idx - r * 48;
    int k = rem / 3, c = rem - k * 3;
    int n = n0 + r;
    float v = Ast[2048 + r * 16 + k] * nesc[r * 4 + 1 + c] +
              Ast[2304 + c * 256 + r * 16 + k];
    A.out_v[n * 48 + rem] = A.x_v[n * 48 + rem] + v;
  }
}

// ---------------------------------------------------------------------------
extern "C" void kernel_launch(void* const* d_in, const int* in_sizes, int n_in,
                              void* d_out, int out_size, void* d_ws, size_t ws_size,
                              hipStream_t stream) {
  (void)out_size; (void)ws_size;
  const float* x_s    = (const float*)d_in[0];
  const float* x_v    = (const float*)d_in[1];
  const float* xc_s   = (const float*)d_in[2];
  const float* xc_v   = (const float*)d_in[3];
  const int*   edge   = (const int*)  d_in[4];
  const int*   il_par = (const int*)  d_in[5];
  const float* ea0    = (const float*)d_in[6];
  const float* eav    = (const float*)d_in[7];
  const float* eila0  = (const float*)d_in[8];
  const float* eilav  = (const float*)d_in[9];
  const float* na0    = (const float*)d_in[10];
  const float* nav    = (const float*)d_in[11];
  const float* amf    = (const float*)d_in[12];
  const float* amf_il = (const float*)d_in[13];

  // Param leaves: sets in order cp1,cp2,m1,m2,u1,u2 (both insertion & sorted).
  // Within a set, detect insertion (Wss,Wvs,Wsv,...) vs jax-sorted
  // (Wss,Wsv,Wvs,...) via the size of leaf #15 (cp1: Wvs=1536 vs Wsv=1040).
  bool jax_sorted = (n_in > 15) && (in_sizes[15] == 65 * 16);
  const float* W[6][6]; // [set][Wss,Wvs,Wsv,Wvv,Wvx,b]
  for (int s = 0; s < 6; ++s) {
    int base = 14 + s * 6;
    W[s][0] = (const float*)d_in[base + 0];
    W[s][1] = (const float*)d_in[base + (jax_sorted ? 2 : 1)];
    W[s][2] = (const float*)d_in[base + (jax_sorted ? 1 : 2)];
    W[s][3] = (const float*)d_in[base + 3];
    W[s][4] = (const float*)d_in[base + 4];
    W[s][5] = (const float*)d_in[base + 5];
  }

  static const int SD[6][4] = {  // s_in, v_in, s_out, v_out
    {65,32,48,16},{32,16,48,16},{65,32,48,16},{32,16,48,16},{64,32,48,16},{32,16,32,16}};

  float* wsf = (float*)d_ws;
  size_t off = 0;
  const float* pW1[6]; const float* pW2[6]; const float* pW3[6];
  for (int s = 0; s < 6; ++s) {
    int si = SD[s][0], vi = SD[s][1], so = SD[s][2], vo = SD[s][3];
    int Kp1 = (si + vi + 3) & ~3, Np1 = (so + 15) & ~15;
    int Kp2 = (si + 3) & ~3,      Np2 = (vo + 15) & ~15;
    int Kp3 = (2*vi + 3) & ~3,    Np3 = (vo + 15) & ~15;
    float* p1 = wsf + off; off += (size_t)Kp1 * Np1;
    float* p2 = wsf + off; off += (size_t)Kp2 * Np2;
    float* p3 = wsf + off; off += (size_t)Kp3 * Np3;
    int t1 = (Np1/16)*(Kp1/4)*32, t2 = (Np2/16)*(Kp2/4)*32, t3 = (Np3/16)*(Kp3/4)*32;
    pack_kernel<<<(t1+255)/256,256,0,stream>>>(W[s][0], W[s][1], si, vi, so, Kp1, Np1, p1);
    pack_kernel<<<(t2+255)/256,256,0,stream>>>(W[s][2], nullptr, si, 0,  vo, Kp2, Np2, p2);
    pack_kernel<<<(t3+255)/256,256,0,stream>>>(W[s][3], W[s][4], vi, vi, vo, Kp3, Np3, p3);
    pW1[s] = p1; pW2[s] = p2; pW3[s] = p3;
  }

  // d_out doubles as the scatter accumulator (identical shape/indexing).
  float* aggs = (float*)d_out;
  float* aggv = (float*)d_out + (size_t)NN * 32;
  int tot = NN * 80;
  zero_kernel<<<(tot + 255) / 256, 256, 0, stream>>>((float*)d_out, tot);

  EdgeArgs ea;
  ea.x_s = x_s; ea.x_v = x_v; ea.edge = edge;
  ea.ea0 = ea0; ea.eav = eav; ea.amf = amf;
  ea.W1a = pW1[2]; ea.W2a = pW2[2]; ea.W3a = pW3[2]; ea.ba = W[2][5];
  ea.W1b = pW1[3]; ea.W2b = pW2[3]; ea.W3b = pW3[3]; ea.bb = W[3][5];
  ea.agg_s = aggs; ea.agg_v = aggv;
  edge_kernel<<<EE / 16, 32, 0, stream>>>(ea);

  NodeArgs na;
  na.x_s = x_s; na.x_v = x_v; na.xc_s = xc_s; na.xc_v = xc_v;
  na.il_parent = il_par;
  na.eila0 = eila0; na.eilav = eilav; na.amf_il = amf_il;
  na.na0 = na0; na.nav = nav;
  na.c1W1 = pW1[0]; na.c1W2 = pW2[0]; na.c1W3 = pW3[0]; na.c1b = W[0][5];
  na.c2W1 = pW1[1]; na.c2W2 = pW2[1]; na.c2W3 = pW3[1]; na.c2b = W[1][5];
  na.u1W1 = pW1[4]; na.u1W2 = pW2[4]; na.u1W3 = pW3[4]; na.u1b = W[4][5];
  na.u2W1 = pW1[5]; na.u2W2 = pW2[5]; na.u2W3 = pW3[5]; na.u2b = W[5][5];
  na.agg_s = aggs; na.agg_v = aggv;
  na.out_s = (float*)d_out; na.out_v = (float*)d_out + (size_t)NN * 32;
  node_kernel<<<NN / 16, 32, 0, stream>>>(na);
}